// Base_64106681860500
// MI455X (gfx1250) — compile-verified
//
#include <hip/hip_runtime.h>
#include <math.h>

typedef __attribute__((ext_vector_type(2))) float v2f;
typedef __attribute__((ext_vector_type(8))) float v8f;
typedef int v4i __attribute__((vector_size(16)));
typedef __attribute__((address_space(1))) v4i* gv4i_p;   // global-space v4i*
typedef __attribute__((address_space(3))) v4i* lv4i_p;   // LDS-space v4i*

#define N_CELLS 16
#define CELL_DIM 16
#define IHH 128
#define IWW 128
#define NPIX (512*512)
#define HID 128

#if defined(__has_builtin)
#if __has_builtin(__builtin_amdgcn_global_load_async_to_lds_b128)
#define HAS_ASYNC_LDS 1
#endif
#endif
#ifndef HAS_ASYNC_LDS
#define HAS_ASYNC_LDS 0
#endif

// fast tanh on the hardware transcendental pipe:
// tanh(x) = sign(x) * (1 - e) / (1 + e),  e = exp(-2|x|).
// ~7 VALU ops (v_exp_f32 + v_rcp_f32 + fma/bfi) vs ~25 for libm tanhf;
// max abs error ~1e-7, monotone, correct saturation for large |x|.
__device__ __forceinline__ float fast_tanh(float v) {
    float a = fabsf(v);
    float e = __expf(-2.f * a);
    float r = (1.f - e) * __frcp_rn(1.f + e);
    return copysignf(r, v);
}

// ---- kernel 1: transpose cells [N][C][IH][IW] -> cellsT [N][IH][IW][C]
// so a bilinear corner gather becomes one contiguous 64B read.
__global__ void transpose_cells(const float* __restrict__ cells,
                                float* __restrict__ cellsT) {
    int idx = blockIdx.x * blockDim.x + threadIdx.x;   // (n,h,w,c), c fastest
    if (idx >= N_CELLS * CELL_DIM * IHH * IWW) return;
    int c = idx & (CELL_DIM - 1);
    int t = idx >> 4;
    int w = t & (IWW - 1); t >>= 7;
    int h = t & (IHH - 1); t >>= 7;
    int n = t;
    cellsT[idx] = cells[((n * CELL_DIM + c) * IHH + h) * IWW + w];
}

// ---- kernel 2: fused multigrid cosine sampling + 3-layer MLP (f32 WMMA)
__global__ __launch_bounds__(256)
void fused_sample_mlp(const float* __restrict__ x,
                      const float* __restrict__ cellsT,
                      const float* __restrict__ W1, const float* __restrict__ b1,
                      const float* __restrict__ W2, const float* __restrict__ b2,
                      const float* __restrict__ W3, const float* __restrict__ b3,
                      float* __restrict__ out)
{
    extern __shared__ float smem[];
    float* sFeat = smem;                  // [256][16]   sampled features (A, layer1)
    float* sW1   = sFeat + 256 * 16;      // [16][128]
    float* sW2   = sW1 + 16 * 128;        // [128][128]
    float* sH1   = sW2 + 128 * 128;       // [256][128]  tanh(layer1) (A, layer2)
    float* sB1   = sH1 + 256 * 128;       // [128]
    float* sB2   = sB1 + 128;             // [128]

    const int tid  = threadIdx.x;
    const int wave = tid >> 5;
    const int lane = tid & 31;
    const int half = lane >> 4;     // 0: lanes 0-15, 1: lanes 16-31
    const int l    = lane & 15;

    // ---- stage weights into LDS (CDNA5 async-to-LDS DMA path, ASYNCcnt)
#if HAS_ASYNC_LDS
    {
        const float4* gW1 = (const float4*)W1;
        const float4* gW2 = (const float4*)W2;
        float4* lW1 = (float4*)sW1;
        float4* lW2 = (float4*)sW2;
        for (int i = tid; i < (16 * 128) / 4; i += 256)
            __builtin_amdgcn_global_load_async_to_lds_b128(
                (gv4i_p)(gW1 + i), (lv4i_p)(lW1 + i), 0, 0);
        for (int i = tid; i < (128 * 128) / 4; i += 256)
            __builtin_amdgcn_global_load_async_to_lds_b128(
                (gv4i_p)(gW2 + i), (lv4i_p)(lW2 + i), 0, 0);
    }
#else
    for (int i = tid; i < (16 * 128) / 4; i += 256)
        ((float4*)sW1)[i] = ((const float4*)W1)[i];
    for (int i = tid; i < (128 * 128) / 4; i += 256)
        ((float4*)sW2)[i] = ((const float4*)W2)[i];
#endif
    if (tid < 128) { sB1[tid] = b1[tid]; sB2[tid] = b2[tid]; }

    // ---- phase A: multigrid cosine-interpolated sampling, 1 pixel / thread
    // (overlaps with the in-flight async weight DMA)
    const int   pix = blockIdx.x * 256 + tid;
    const float PI  = 3.14159265358979323846f;
    float feat[CELL_DIM];
    #pragma unroll
    for (int c = 0; c < CELL_DIM; ++c) feat[c] = 0.f;

    const float gx = (x[2 * pix + 0] + 1.f) * 0.5f * (IWW - 1);
    const float gy = (x[2 * pix + 1] + 1.f) * 0.5f * (IHH - 1);

    for (int n = 0; n < N_CELLS; ++n) {
        const float off = (float)n * (1.f / 16.f);       // multigrid stagger
        float fxp = gx + off, fyp = gy + off;
        float x0f = floorf(fxp), y0f = floorf(fyp);
        float fx = fxp - x0f, fy = fyp - y0f;
        float wx = 0.5f * (1.f - __cosf(PI * fx));       // cosine step weights
        float wy = 0.5f * (1.f - __cosf(PI * fy));
        int x0 = (int)x0f, y0 = (int)y0f;
        const float cw[4]  = {(1.f - wx) * (1.f - wy), wx * (1.f - wy),
                              (1.f - wx) * wy,         wx * wy};
        const int   cxs[4] = {x0, x0 + 1, x0,     x0 + 1};
        const int   cys[4] = {y0, y0,     y0 + 1, y0 + 1};
        const float* gbase = cellsT + (size_t)n * IHH * IWW * CELL_DIM;
        #pragma unroll
        for (int k = 0; k < 4; ++k) {
            int cx = cxs[k], cy = cys[k];
            if (cx >= 0 && cx < IWW && cy >= 0 && cy < IHH) {   // zeros padding
                const float4* v4 =
                    (const float4*)(gbase + ((size_t)cy * IWW + cx) * CELL_DIM);
                float wgt = cw[k];
                #pragma unroll
                for (int q = 0; q < 4; ++q) {
                    float4 v = v4[q];
                    feat[4 * q + 0] += wgt * v.x; feat[4 * q + 1] += wgt * v.y;
                    feat[4 * q + 2] += wgt * v.z; feat[4 * q + 3] += wgt * v.w;
                }
            }
        }
    }
    #pragma unroll
    for (int c = 0; c < CELL_DIM; ++c) sFeat[tid * CELL_DIM + c] = feat[c];

#if HAS_ASYNC_LDS
    asm volatile("s_wait_asynccnt 0" ::: "memory");  // drain async W1/W2 copies
#endif
    __syncthreads();

    // Wave w owns M-tiles mt0 = 2w (rows 32w..32w+15) and mt1 = 2w+1.
    const int mt0 = 2 * wave;
    const int mt1 = 2 * wave + 1;

    // ---- phase B: layer1  h1 = tanh(feat[256,16] @ W1[16,128] + b1)
    // Both M-tiles kept live so each W1 B-fragment feeds two WMMAs.
    {
        v8f acc0[8], acc1[8];
        #pragma unroll
        for (int nt = 0; nt < 8; ++nt) {
            float bb = sB1[16 * nt + l];              // bias broadcast per column
            #pragma unroll
            for (int e = 0; e < 8; ++e) { acc0[nt][e] = bb; acc1[nt][e] = bb; }
        }
        #pragma unroll
        for (int kt = 0; kt < 4; ++kt) {              // K = 16 -> 4 steps of 4
            const int kk = 4 * kt + 2 * half;
            v2f a0, a1;                               // A[m=l][k=kk, kk+1]
            a0.x = sFeat[(16 * mt0 + l) * CELL_DIM + kk + 0];
            a0.y = sFeat[(16 * mt0 + l) * CELL_DIM + kk + 1];
            a1.x = sFeat[(16 * mt1 + l) * CELL_DIM + kk + 0];
            a1.y = sFeat[(16 * mt1 + l) * CELL_DIM + kk + 1];
            #pragma unroll
            for (int nt = 0; nt < 8; ++nt) {
                v2f b;                                // B[k][n=16*nt+l]
                b.x = sW1[(kk + 0) * HID + 16 * nt + l];
                b.y = sW1[(kk + 1) * HID + 16 * nt + l];
                acc0[nt] = __builtin_amdgcn_wmma_f32_16x16x4_f32(
                    false, a0, false, b, (short)0, acc0[nt], false, false);
                acc1[nt] = __builtin_amdgcn_wmma_f32_16x16x4_f32(
                    false, a1, false, b, (short)0, acc1[nt], false, false);
            }
        }
        #pragma unroll
        for (int nt = 0; nt < 8; ++nt)
            #pragma unroll
            for (int v = 0; v < 8; ++v) {             // C row = v + 8*half
                sH1[(16 * mt0 + 8 * half + v) * HID + 16 * nt + l] =
                    fast_tanh(acc0[nt][v]);
                sH1[(16 * mt1 + 8 * half + v) * HID + 16 * nt + l] =
                    fast_tanh(acc1[nt][v]);
            }
    }

    __syncthreads();

    // ---- phase C: layer2 (WMMA, B-fragment shared by both M-tiles)
    //               + fused layer3 per-row dot product
    float w3l[8];
    #pragma unroll
    for (int nt = 0; nt < 8; ++nt) w3l[nt] = W3[16 * nt + l];
    const float bias3 = b3[0];

    {
        v8f acc0[8], acc1[8];
        #pragma unroll
        for (int nt = 0; nt < 8; ++nt) {
            float bb = sB2[16 * nt + l];
            #pragma unroll
            for (int e = 0; e < 8; ++e) { acc0[nt][e] = bb; acc1[nt][e] = bb; }
        }
        const int rowA0 = (16 * mt0 + l) * HID;
        const int rowA1 = (16 * mt1 + l) * HID;
        for (int kt = 0; kt < 32; ++kt) {             // K = 128 -> 32 steps of 4
            const int kk = 4 * kt + 2 * half;
            v2f a0, a1;
            a0.x = sH1[rowA0 + kk + 0];
            a0.y = sH1[rowA0 + kk + 1];
            a1.x = sH1[rowA1 + kk + 0];
            a1.y = sH1[rowA1 + kk + 1];
            #pragma unroll
            for (int nt = 0; nt < 8; ++nt) {
                v2f b;
                b.x = sW2[(kk + 0) * HID + 16 * nt + l];
                b.y = sW2[(kk + 1) * HID + 16 * nt + l];
                acc0[nt] = __builtin_amdgcn_wmma_f32_16x16x4_f32(
                    false, a0, false, b, (short)0, acc0[nt], false, false);
                acc1[nt] = __builtin_amdgcn_wmma_f32_16x16x4_f32(
                    false, a1, false, b, (short)0, acc1[nt], false, false);
            }
        }
        // layer3: out[row] = sum_c tanh(h2[row][c]) * W3[c] + b3
        #pragma unroll
        for (int v = 0; v < 8; ++v) {
            float s0 = 0.f, s1 = 0.f;
            #pragma unroll
            for (int nt = 0; nt < 8; ++nt) {
                s0 += fast_tanh(acc0[nt][v]) * w3l[nt];
                s1 += fast_tanh(acc1[nt][v]) * w3l[nt];
            }
            #pragma unroll
            for (int m = 8; m >= 1; m >>= 1) {
                s0 += __shfl_xor(s0, m, 16);
                s1 += __shfl_xor(s1, m, 16);
            }
            if (l == 0) {
                out[blockIdx.x * 256 + 16 * mt0 + 8 * half + v] = s0 + bias3;
                out[blockIdx.x * 256 + 16 * mt1 + 8 * half + v] = s1 + bias3;
            }
        }
    }
}

extern "C" void kernel_launch(void* const* d_in, const int* in_sizes, int n_in,
                              void* d_out, int out_size, void* d_ws, size_t ws_size,
                              hipStream_t stream) {
    const float* x     = (const float*)d_in[0];
    const float* cells = (const float*)d_in[1];
    const float* W1    = (const float*)d_in[2];
    const float* b1    = (const float*)d_in[3];
    const float* W2    = (const float*)d_in[4];
    const float* b2    = (const float*)d_in[5];
    const float* W3    = (const float*)d_in[6];
    const float* b3    = (const float*)d_in[7];
    float* out    = (float*)d_out;
    float* cellsT = (float*)d_ws;   // 16 MB: [N][IH][IW][C]

    const int telems = N_CELLS * CELL_DIM * IHH * IWW;
    transpose_cells<<<telems / 256, 256, 0, stream>>>(cells, cellsT);

    const size_t shmem =
        (256 * 16 + 16 * 128 + 128 * 128 + 256 * 128 + 256) * sizeof(float);
    fused_sample_mlp<<<NPIX / 256, 256, shmem, stream>>>(
        x, cellsT, W1, b1, W2, b2, W3, b3, out);
}